// MultiDomainMultiCriteriaClassifier_68350109548839
// MI455X (gfx1250) — compile-verified
//
#include <hip/hip_runtime.h>
#include <math.h>

// Problem constants (match reference): P=8, MAXC=16, D=768, H=8, B=65536
#define DD     768
#define HH     8
#define MAXCC  16
#define PP     8
#define WSTR   772   // LDS row stride for W (bank-conflict-free ds_load_b64)

typedef __attribute__((ext_vector_type(2))) float v2f;
typedef __attribute__((ext_vector_type(8))) float v8f;

// ---------------------------------------------------------------------------
// Kernel 1: critb[p*16+j] = dot(criteria[p][j], W[head_map[p][j]]) + b[head]
// 128 independent length-768 dot products; one wave32 each.
// ---------------------------------------------------------------------------
__global__ __launch_bounds__(32)
void crit_kernel(const float* __restrict__ criteria,
                 const float* __restrict__ W,
                 const float* __restrict__ b,
                 const int*   __restrict__ head_map,
                 float*       __restrict__ critb)
{
    const int pj   = blockIdx.x;          // 0..127  (p = pj/16, j = pj%16)
    const int head = head_map[pj];
    const float* __restrict__ crow = criteria + (size_t)pj * DD;
    const float* __restrict__ wrow = W + (size_t)head * DD;

    const int lane = threadIdx.x;
    float acc = 0.0f;
    #pragma unroll 4
    for (int k = lane; k < DD; k += 32)
        acc = fmaf(crow[k], wrow[k], acc);

    // wave32 reduction
    #pragma unroll
    for (int off = 16; off > 0; off >>= 1)
        acc += __shfl_down(acc, off, 32);

    if (lane == 0)
        critb[pj] = acc + b[head];
}

// ---------------------------------------------------------------------------
// Kernel 2: per-wave 16-row tile of text_logits via V_WMMA_F32_16X16X4_F32,
// then head gather + sigmoid + mask epilogue.
// Block = 256 threads = 8 waves = 128 rows. Grid = B/128.
// ---------------------------------------------------------------------------
__global__ __launch_bounds__(256)
void MultiDomainMultiCriteriaClassifier_68350109548839_kernel(
        const float* __restrict__ emb,       // [B, D]
        const float* __restrict__ W,         // [H, D]
        const float* __restrict__ critb,     // [P*MAXC] = crit_logits + b[head]
        const int*   __restrict__ pidx,      // [B]
        const int*   __restrict__ head_map,  // [P, MAXC]
        const int*   __restrict__ lengths,   // [P]
        float*       __restrict__ out,       // [B, MAXC]
        float*       __restrict__ maskout,   // [B, MAXC]
        int Bn)
{
    // W staged once per block, zero-padded to 16 rows (heads 8..15 = 0 so the
    // unused N-half of the WMMA tile accumulates zeros). Stride 772 floats:
    // 3088 B between rows => 16 lanes map to distinct bank pairs.
    __shared__ float lds_W[16 * WSTR];            // 49,408 B
    __shared__ float lds_logits[8 * 16 * 16];     //  8,192 B (per-wave 16x16)

    const int tid = threadIdx.x;
    for (int i = tid; i < 16 * DD; i += 256) {
        const int r = i / DD, c = i - r * DD;
        lds_W[r * WSTR + c] = (r < HH) ? W[r * DD + c] : 0.0f;
    }
    __syncthreads();

    const int wave = tid >> 5;
    const int lane = tid & 31;
    const int half = lane >> 4;    // selects K pair {0,1} vs {2,3}
    const int l16  = lane & 15;    // A: row-in-tile, B: head (N)

    const int rowBase = blockIdx.x * 128 + wave * 16;
    if (rowBase >= Bn) return;

    // A fragment source: lane l -> emb[rowBase + (l&15)][k + 2*(l>>4) .. +1]
    const float* __restrict__ arow =
        emb + (size_t)(rowBase + l16) * DD + (size_t)(half * 2);
    // B fragment source (W^T): lane l -> W[l&15][k + 2*(l>>4) .. +1] from LDS
    const float* brow = lds_W + l16 * WSTR + half * 2;

    v8f acc = {};
    #pragma unroll 8
    for (int k = 0; k < DD; k += 4) {
        v2f a  = *(const v2f*)(arow + k);   // global_load_b64
        v2f bf = *(const v2f*)(brow + k);   // ds_load_b64
        // D = A(16x4,f32) x B(4x16,f32) + C(16x16,f32)
        acc = __builtin_amdgcn_wmma_f32_16x16x4_f32(
                  /*neg_a=*/false, a, /*neg_b=*/false, bf,
                  /*c_mod=*/(short)0, acc,
                  /*reuse_a=*/false, /*reuse_b=*/false);
    }

    // Spill accumulator tile to this wave's LDS scratch:
    // C layout: VGPR r, lane l => M = r + 8*(l>>4), N = l&15
    float* __restrict__ lg = lds_logits + wave * 256;
    #pragma unroll
    for (int r = 0; r < 8; ++r)
        lg[(r + half * 8) * 16 + l16] = acc[r];
    __syncthreads();

    // Epilogue: 256 outputs per tile; lane (half,l16) covers (row=2t+half, j=l16)
    #pragma unroll
    for (int t = 0; t < 8; ++t) {
        const int r    = t * 2 + half;       // row within tile
        const int j    = l16;                // criterion slot
        const int grow = rowBase + r;        // global row
        const int pid  = pidx[grow];
        const int hm   = head_map[pid * MAXCC + j];
        const float logit = lg[r * 16 + hm] + critb[pid * MAXCC + j];
        const float m = (j < lengths[pid]) ? 1.0f : 0.0f;
        const float s = 1.0f / (1.0f + __expf(-logit));
        out    [(size_t)grow * MAXCC + j] = s * m;   // coalesced 128B/iter
        maskout[(size_t)grow * MAXCC + j] = m;
    }
}

// ---------------------------------------------------------------------------
extern "C" void kernel_launch(void* const* d_in, const int* in_sizes, int n_in,
                              void* d_out, int out_size, void* d_ws, size_t ws_size,
                              hipStream_t stream)
{
    const float* emb      = (const float*)d_in[0];   // [B, D]
    const float* criteria = (const float*)d_in[1];   // [P, MAXC, D]
    const float* W        = (const float*)d_in[2];   // [H, D]
    const float* b        = (const float*)d_in[3];   // [H]
    const int*   pidx     = (const int*)  d_in[4];   // [B]
    const int*   head_map = (const int*)  d_in[5];   // [P, MAXC]
    const int*   lengths  = (const int*)  d_in[6];   // [P]

    const int Bn = in_sizes[4];                      // 65536
    float* out     = (float*)d_out;                  // padded_outputs [B, MAXC]
    float* maskout = out + (size_t)Bn * MAXCC;       // mask           [B, MAXC]
    float* critb   = (float*)d_ws;                   // 128 floats scratch

    crit_kernel<<<PP * MAXCC, 32, 0, stream>>>(criteria, W, b, head_map, critb);

    const int grid = (Bn + 127) / 128;               // 512 blocks
    MultiDomainMultiCriteriaClassifier_68350109548839_kernel<<<grid, 256, 0, stream>>>(
        emb, W, critb, pidx, head_map, lengths, out, maskout, Bn);
}